// EncFCSwitch_18863496364233
// MI455X (gfx1250) — compile-verified
//
#include <hip/hip_runtime.h>
#include <hip/hip_bf16.h>
#include <math.h>

typedef float v2f __attribute__((ext_vector_type(2)));
typedef float v8f __attribute__((ext_vector_type(8)));
typedef int   v4i_vs __attribute__((vector_size(4 * sizeof(int))));

#define AS1 __attribute__((address_space(1)))
#define AS3 __attribute__((address_space(3)))

#define B_ROWS 16384
#define D_DIM  1024
#define N_BR   8
#define S_DIM  64
#define NS     512   // N_BR * S_DIM
#define KC     64    // K chunk
#define KP     (KC / 2)
#define MT     64    // rows per block
#define AROW   162   // pair-interleaved A row stride (2*64 + 34)
#define BROW   1056  // pair-interleaved B row stride (2*512 + 32)
#define BROW2  544   // pair-interleaved k2 B row stride (2*256 + 32)

#if defined(__has_builtin)
#if __has_builtin(__builtin_amdgcn_global_store_async_from_lds_b128)
#define HAVE_ASYNC_ST 1
#endif
#if __has_builtin(__builtin_amdgcn_s_wait_asynccnt)
#define HAVE_WAIT_ASYNC 1
#endif
#endif

// ---------------- WMMA wrapper (f32 full-precision matrix path) -------------
static __device__ __forceinline__ v8f wmma_f32(v2f a, v2f b, v8f c) {
  // (neg_a, A, neg_b, B, c_mod, C, reuse_a, reuse_b)
  return __builtin_amdgcn_wmma_f32_16x16x4_f32(false, a, false, b, (short)0, c,
                                               false, false);
}

// ---------------- exact JAX threefry2x32 Gumbel noise -----------------------
static __device__ __forceinline__ unsigned rotl32(unsigned x, int r) {
  return (x << r) | (x >> (32 - r));
}
static __device__ __forceinline__ void threefry2x32(unsigned k0, unsigned k1,
                                                    unsigned x0, unsigned x1,
                                                    unsigned& o0, unsigned& o1) {
  unsigned ks2 = k0 ^ k1 ^ 0x1BD11BDAu;
  x0 += k0; x1 += k1;
  const int rA[4] = {13, 15, 26, 6};
  const int rB[4] = {17, 29, 16, 24};
#pragma unroll
  for (int i = 0; i < 4; ++i) { x0 += x1; x1 = rotl32(x1, rA[i]); x1 ^= x0; }
  x0 += k1;  x1 += ks2 + 1u;
#pragma unroll
  for (int i = 0; i < 4; ++i) { x0 += x1; x1 = rotl32(x1, rB[i]); x1 ^= x0; }
  x0 += ks2; x1 += k0 + 2u;
#pragma unroll
  for (int i = 0; i < 4; ++i) { x0 += x1; x1 = rotl32(x1, rA[i]); x1 ^= x0; }
  x0 += k0;  x1 += k1 + 3u;
#pragma unroll
  for (int i = 0; i < 4; ++i) { x0 += x1; x1 = rotl32(x1, rB[i]); x1 ^= x0; }
  x0 += k1;  x1 += ks2 + 4u;
#pragma unroll
  for (int i = 0; i < 4; ++i) { x0 += x1; x1 = rotl32(x1, rA[i]); x1 ^= x0; }
  x0 += ks2; x1 += k0 + 5u;
  o0 = x0; o1 = x1;
}

// gumbel for flat element j of a [B,N] f32 draw with jax.random.key(1234)
static __device__ __forceinline__ float jax_gumbel(unsigned j) {
  const unsigned halfN = (B_ROWS * N_BR) / 2;  // 65536
  unsigned x0 = (j < halfN) ? j : (j - halfN);
  unsigned x1 = x0 + halfN;
  unsigned o0, o1;
  threefry2x32(0u, 1234u, x0, x1, o0, o1);
  unsigned bits = (j < halfN) ? o0 : o1;
  unsigned fb = (bits >> 9) | 0x3f800000u;          // [1,2)
  float u = __uint_as_float(fb) - 1.0f;             // [0,1)
  float e = -log1pf(-u);                            // exponential sample
  return -logf(e + 1e-20f);                         // gumbel
}

// ---------------- Kernel 0: zero per-expert counters ------------------------
__global__ void k_zero(int* counts) {
  if (threadIdx.x < N_BR) counts[threadIdx.x] = 0;
}

// ---------------- Kernel 1: h GEMM + logits + gumbel argmax + dispatch ------
struct K1Smem {
  union {
    float Bp[KP * BROW];    // pair-interleaved W_in chunk (~132 KB)
    float h_all[MT * NS];   // full h tile afterwards (128 KB)
  };
  float A[KP * AROW];       // pair-interleaved transposed relu(x) (~20.3 KB)
  float logits[MT * N_BR];
  int   idx[MT];
};

__global__ __launch_bounds__(256) void k1(
    const float* __restrict__ x, const float* __restrict__ W_in,
    const float* __restrict__ b_in, const float* __restrict__ W_sw,
    const float* __restrict__ b_sw,
    float* __restrict__ out_logits, float* __restrict__ out_index,
    float* __restrict__ out_hard,
    float* __restrict__ h_sel, float* __restrict__ gate_ws,
    int* __restrict__ branch_rows, int* __restrict__ counts) {
  __shared__ K1Smem sm;
  const int t = threadIdx.x;
  const int wave = t >> 5;
  const int lane = t & 31;
  const int half = lane >> 4;
  const int l15 = lane & 15;
  const int rowBase = blockIdx.x * MT;
  const int br = wave;          // 8 waves == 8 branches (S==64 columns each)
  const int c0 = br * S_DIM;

  v8f acc[4][4];
#pragma unroll
  for (int mt = 0; mt < 4; ++mt)
#pragma unroll
    for (int nt = 0; nt < 4; ++nt) acc[mt][nt] = (v8f)0.0f;

  for (int k0 = 0; k0 < D_DIM; k0 += KC) {
    // ---- stage relu(x), pair-interleaved transpose: A[(k>>1)][2m+(k&1)] ----
#pragma unroll
    for (int i = 0; i < (MT * KC) / (256 * 4); ++i) {   // 4 iters
      int f = (t + 256 * i) * 4;
      int m = f >> 6, k = f & 63;                        // k multiple of 4
      float4 v = *(const float4*)(x + (size_t)(rowBase + m) * D_DIM + k0 + k);
      int p = k >> 1;                                    // even
      v2f lo; lo.x = fmaxf(v.x, 0.0f); lo.y = fmaxf(v.y, 0.0f);
      v2f hi; hi.x = fmaxf(v.z, 0.0f); hi.y = fmaxf(v.w, 0.0f);
      *(v2f*)(sm.A + (p + 0) * AROW + 2 * m) = lo;
      *(v2f*)(sm.A + (p + 1) * AROW + 2 * m) = hi;
    }
    // ---- stage W_in chunk, pair-interleaved: Bp[p][2c+(k&1)] ----
#pragma unroll
    for (int i = 0; i < (KP * NS) / (256 * 4); ++i) {   // 16 iters
      int e = t + 256 * i;
      int c4 = e & 127, p = e >> 7;                      // c = 4*c4
      int c = c4 * 4;
      int n = c >> 6, s = c & 63;
      const float* g = W_in + (size_t)n * D_DIM * S_DIM +
                       (size_t)(k0 + 2 * p) * S_DIM + s;
      float4 g0 = *(const float4*)(g);
      float4 g1 = *(const float4*)(g + S_DIM);
      float4 o0 = make_float4(g0.x, g1.x, g0.y, g1.y);
      float4 o1 = make_float4(g0.z, g1.z, g0.w, g1.w);
      float* d = sm.Bp + p * BROW + 8 * c4;              // 16B aligned
      *(float4*)(d + 0) = o0;
      *(float4*)(d + 4) = o1;
    }
    // hint: prefetch next W_in chunk while we compute this one
    if (k0 + KC < D_DIM) {
      int n = t & 7, k = (t >> 3) & 31;
      __builtin_prefetch(W_in + (size_t)n * D_DIM * S_DIM +
                             (size_t)(k0 + KC + k * 2) * S_DIM, 0, 1);
    }
    __syncthreads();

    // ---- WMMA over this K chunk: fragments are single ds_load_b64 ----
    for (int kk = 0; kk < KC; kk += 4) {
      const int p = (kk >> 1) + half;  // lanes0-15: K=kk,kk+1; 16-31: kk+2,kk+3
      v2f afr[4], bfr[4];
#pragma unroll
      for (int mt = 0; mt < 4; ++mt)
        afr[mt] = *(const v2f*)(sm.A + p * AROW + 2 * (mt * 16 + l15));
#pragma unroll
      for (int nt = 0; nt < 4; ++nt)
        bfr[nt] = *(const v2f*)(sm.Bp + p * BROW + 2 * (c0 + nt * 16 + l15));
#pragma unroll
      for (int mt = 0; mt < 4; ++mt)
#pragma unroll
        for (int nt = 0; nt < 4; ++nt)
          acc[mt][nt] = wmma_f32(afr[mt], bfr[nt], acc[mt][nt]);
    }
    __syncthreads();
  }

  // ---- epilogue: bias, logits partials, spill h to LDS (reuses Bp) ----
  float wsw[4], bia[4];
#pragma unroll
  for (int nt = 0; nt < 4; ++nt) {
    int s = nt * 16 + l15;
    wsw[nt] = W_sw[br * S_DIM + s];
    bia[nt] = b_in[br * S_DIM + s];
  }
  const float bsw = b_sw[br];
#pragma unroll
  for (int mt = 0; mt < 4; ++mt) {
#pragma unroll
    for (int r = 0; r < 8; ++r) {
      const int m = mt * 16 + r + 8 * half;
      float lacc = 0.0f;
#pragma unroll
      for (int nt = 0; nt < 4; ++nt) {
        float h = acc[mt][nt][r] + bia[nt];
        lacc += h * wsw[nt];
        sm.h_all[m * NS + c0 + nt * 16 + l15] = h;
      }
      // reduce 16-lane halves (butterfly stays inside each half for mask<16)
      lacc += __shfl_xor(lacc, 1);
      lacc += __shfl_xor(lacc, 2);
      lacc += __shfl_xor(lacc, 4);
      lacc += __shfl_xor(lacc, 8);
      if (l15 == 0) sm.logits[m * N_BR + br] = lacc + bsw;
    }
  }
  __syncthreads();

  // ---- gumbel-softmax sample per row (one thread per row) ----
  if (t < MT) {
    const int b = rowBase + t;
    float g[N_BR], ysoft[N_BR];
    float mx = -INFINITY;
#pragma unroll
    for (int n = 0; n < N_BR; ++n) {
      float logit = sm.logits[t * N_BR + n];
      out_logits[(size_t)b * N_BR + n] = logit;
      g[n] = logit + jax_gumbel((unsigned)(b * N_BR + n));  // TAU == 1
      mx = fmaxf(mx, g[n]);
    }
    float sum = 0.0f;
#pragma unroll
    for (int n = 0; n < N_BR; ++n) { ysoft[n] = expf(g[n] - mx); sum += ysoft[n]; }
    float inv = 1.0f / sum;
    int idx = 0; float best = ysoft[0] * inv; ysoft[0] = best;
#pragma unroll
    for (int n = 1; n < N_BR; ++n) {
      ysoft[n] *= inv;
      if (ysoft[n] > best) { best = ysoft[n]; idx = n; }  // first-max wins
    }
    float gate = 1.0f;
#pragma unroll
    for (int n = 0; n < N_BR; ++n) {
      float oh = (n == idx) ? 1.0f : 0.0f;
      float yh = oh - ysoft[n] + ysoft[n];   // straight-through value path
      out_hard[(size_t)b * N_BR + n] = yh;
      if (n == idx) gate = yh;
    }
    out_index[b] = (float)idx;
    sm.idx[t] = idx;
    gate_ws[b] = gate;
    int pos = atomicAdd(&counts[idx], 1);
    branch_rows[idx * B_ROWS + pos] = b;
  }
  __syncthreads();

  // ---- gather selected branch's hidden vector to ws (async LDS->global) ----
#pragma unroll
  for (int i = 0; i < (MT * S_DIM) / (256 * 4); ++i) {   // 4 iters of b128
    int e = t + 256 * i;
    int m = e >> 4, s4 = (e & 15) * 4;
    int lidx = m * NS + sm.idx[m] * S_DIM + s4;          // 16B aligned
    size_t gidx = (size_t)(rowBase + m) * S_DIM + s4;    // 16B aligned
#if defined(HAVE_ASYNC_ST)
    __builtin_amdgcn_global_store_async_from_lds_b128(
        (AS1 v4i_vs*)(h_sel + gidx),
        (AS3 v4i_vs*)(sm.h_all + lidx), 0, 0);
#else
    *(float4*)(h_sel + gidx) = *(const float4*)(sm.h_all + lidx);
#endif
  }
#if defined(HAVE_ASYNC_ST) && defined(HAVE_WAIT_ASYNC)
  __builtin_amdgcn_s_wait_asynccnt(0);
#endif
}

// ---------------- Kernel 2: per-expert [rows x 64] x [64 x 1024] GEMM -------
struct K2Smem {
  float A[KP * AROW];       // pair-interleaved h_sel tile (~20.3 KB)
  float Bp[KP * BROW2];     // pair-interleaved W_out 256-col slab (~68 KB)
  int   rowA[MT];           // clamped row (for loads)
  int   rowV[MT];           // -1 when padding (skip stores)
  float gate[MT];
};

__global__ __launch_bounds__(256) void k2(
    const float* __restrict__ x, const float* __restrict__ W_out,
    const float* __restrict__ b_out, const float* __restrict__ h_sel,
    const float* __restrict__ gate_ws, const int* __restrict__ branch_rows,
    const int* __restrict__ counts, float* __restrict__ out) {
  const int br = blockIdx.y;
  const int tile = blockIdx.x;
  const int cnt = counts[br];
  if (tile * MT >= cnt) return;   // uniform early exit

  __shared__ K2Smem sm;
  const int t = threadIdx.x;
  const int wave = t >> 5;
  const int lane = t & 31;
  const int half = lane >> 4;
  const int l15 = lane & 15;

  if (t < MT) {
    int li = tile * MT + t;
    int cl = (li < cnt) ? li : cnt - 1;
    int row = branch_rows[br * B_ROWS + cl];
    sm.rowA[t] = row;
    sm.rowV[t] = (li < cnt) ? row : -1;
    sm.gate[t] = gate_ws[row];
  }
  __syncthreads();
#pragma unroll
  for (int i = 0; i < (MT * S_DIM) / (256 * 4); ++i) {   // 4 iters
    int f = (t + 256 * i) * 4;
    int m = f >> 6, k = f & 63;
    float4 v = *(const float4*)(h_sel + (size_t)sm.rowA[m] * S_DIM + k);
    int p = k >> 1;
    v2f lo; lo.x = v.x; lo.y = v.y;
    v2f hi; hi.x = v.z; hi.y = v.w;
    *(v2f*)(sm.A + (p + 0) * AROW + 2 * m) = lo;
    *(v2f*)(sm.A + (p + 1) * AROW + 2 * m) = hi;
  }

  const float* Wg = W_out + (size_t)br * S_DIM * D_DIM;
  for (int j = 0; j < 4; ++j) {                 // 256-column slabs
    __syncthreads();                            // protect previous slab reads
    // ---- stage W_out slab, pair-interleaved ----
#pragma unroll
    for (int i = 0; i < (KP * 256) / (256 * 4); ++i) {   // 8 iters
      int e = t + 256 * i;
      int c4 = e & 63, p = e >> 6;
      const float* g = Wg + (size_t)(2 * p) * D_DIM + j * 256 + 4 * c4;
      float4 g0 = *(const float4*)(g);
      float4 g1 = *(const float4*)(g + D_DIM);
      float4 o0 = make_float4(g0.x, g1.x, g0.y, g1.y);
      float4 o1 = make_float4(g0.z, g1.z, g0.w, g1.w);
      float* d = sm.Bp + p * BROW2 + 8 * c4;
      *(float4*)(d + 0) = o0;
      *(float4*)(d + 4) = o1;
    }
    __syncthreads();

    const int cw = wave * 32;                   // wave's cols within slab
    v8f acc[4][2];
#pragma unroll
    for (int mt = 0; mt < 4; ++mt)
#pragma unroll
      for (int nt = 0; nt < 2; ++nt) acc[mt][nt] = (v8f)0.0f;

    for (int kk = 0; kk < S_DIM; kk += 4) {
      const int p = (kk >> 1) + half;
      v2f afr[4], bfr[2];
#pragma unroll
      for (int mt = 0; mt < 4; ++mt)
        afr[mt] = *(const v2f*)(sm.A + p * AROW + 2 * (mt * 16 + l15));
#pragma unroll
      for (int nt = 0; nt < 2; ++nt)
        bfr[nt] = *(const v2f*)(sm.Bp + p * BROW2 + 2 * (cw + nt * 16 + l15));
#pragma unroll
      for (int mt = 0; mt < 4; ++mt)
#pragma unroll
        for (int nt = 0; nt < 2; ++nt)
          acc[mt][nt] = wmma_f32(afr[mt], bfr[nt], acc[mt][nt]);
    }

    float bo[2];
#pragma unroll
    for (int nt = 0; nt < 2; ++nt)
      bo[nt] = b_out[(size_t)br * D_DIM + j * 256 + cw + nt * 16 + l15];
#pragma unroll
    for (int mt = 0; mt < 4; ++mt)
#pragma unroll
      for (int r = 0; r < 8; ++r) {
        const int m = mt * 16 + r + 8 * half;
        const int row = sm.rowV[m];
        if (row >= 0) {
          const float g = sm.gate[m];
#pragma unroll
          for (int nt = 0; nt < 2; ++nt) {
            int c = j * 256 + cw + nt * 16 + l15;
            out[(size_t)row * D_DIM + c] =
                x[(size_t)row * D_DIM + c] + (acc[mt][nt][r] + bo[nt]) * g;
          }
        }
      }
  }
}

// ---------------- launch ----------------------------------------------------
extern "C" void kernel_launch(void* const* d_in, const int* in_sizes, int n_in,
                              void* d_out, int out_size, void* d_ws,
                              size_t ws_size, hipStream_t stream) {
  const float* x     = (const float*)d_in[0];
  const float* W_in  = (const float*)d_in[1];
  const float* b_in  = (const float*)d_in[2];
  const float* W_sw  = (const float*)d_in[3];
  const float* b_sw  = (const float*)d_in[4];
  const float* W_out = (const float*)d_in[5];
  const float* b_out = (const float*)d_in[6];

  float* out = (float*)d_out;
  float* out_main   = out;
  float* out_logits = out + (size_t)B_ROWS * D_DIM;
  float* out_index  = out_logits + (size_t)B_ROWS * N_BR;
  float* out_hard   = out_index + B_ROWS;

  // workspace layout
  float* h_sel   = (float*)d_ws;                            // B*64 f32
  float* gate_ws = h_sel + (size_t)B_ROWS * S_DIM;          // B f32
  int* branch_rows = (int*)(gate_ws + B_ROWS);              // 8*B int
  int* counts      = branch_rows + (size_t)N_BR * B_ROWS;   // 8 int

  k_zero<<<1, 32, 0, stream>>>(counts);
  k1<<<B_ROWS / MT, 256, 0, stream>>>(x, W_in, b_in, W_sw, b_sw, out_logits,
                                      out_index, out_hard, h_sel, gate_ws,
                                      branch_rows, counts);
  k2<<<dim3(B_ROWS / MT, N_BR), 256, 0, stream>>>(
      x, W_out, b_out, h_sel, gate_ws, branch_rows, counts, out_main);
  (void)in_sizes; (void)n_in; (void)out_size; (void)ws_size;
}